// FocalLoss_63565515981097
// MI455X (gfx1250) — compile-verified
//
#include <hip/hip_runtime.h>
#include <hip/hip_bf16.h>

// Problem constants (match reference)
constexpr int   kB = 8, kA = 100000, kC = 80, kM = 50, kH = 512, kW = 512;
constexpr float kMaxPos2  = 25.0f;      // MAX_POS^2
constexpr float kNegPos2  = 56.25f;     // (1.5*MAX_POS)^2
constexpr float kMaxAng   = 10.0f;
constexpr float kNegAng   = 15.0f;
constexpr float kDamp     = 0.1f;

typedef __attribute__((ext_vector_type(2))) float v2f;
typedef __attribute__((ext_vector_type(8))) float v8f;

// ---------------------------------------------------------------------------
// Kernel 1: anchor->annotation assignment via V_WMMA_F32_16X16X4_F32.
// d2(m,n) = [1, mx, my, |m|^2] . [|a|^2, -2ax, -2ay, 1]^T  (K=4 GEMM)
// One wave handles 16 anchors vs 64 padded annotations (4 WMMAs).
// Block = 256 threads (8 waves = 128 anchors). grid = (ceil(A/128), B).
// Annotation fragments are built branchlessly from a clamped float4 load so
// the four WMMAs issue without exec-mask traffic between them.
// ---------------------------------------------------------------------------
__global__ __launch_bounds__(256)
void assign_wmma_kernel(const float* __restrict__ anchors,      // (A,3)
                        const float* __restrict__ annotations,  // (B,M,4)
                        const float* __restrict__ regressions,  // (B,A,3)
                        const unsigned char* __restrict__ states,// (B,1,H,W) bool
                        int*   __restrict__ codes,              // (B,A)
                        float* __restrict__ damps,              // (B,A)
                        float* __restrict__ xyPart,             // (B,nb1)
                        float* __restrict__ angPart,            // (B,nb1)
                        float* __restrict__ nposPart,           // (B,nb1)
                        int nb1)
{
    const int b    = blockIdx.y;
    const int lane = threadIdx.x & 31;
    const int wave = threadIdx.x >> 5;
    const int half = lane >> 4;     // 0 = lanes 0-15, 1 = lanes 16-31
    const int l15  = lane & 15;
    const int anchorBase = (blockIdx.x * 8 + wave) * 16;

    float xyC = 0.0f, angC = 0.0f, nposC = 0.0f;

    if (anchorBase < kA) {          // wave-uniform: EXEC all-ones inside
        // ---- B fragment (4x16, K x N). Column n = l15 -> anchor anchorBase+l15.
        // Layout: VGPR0/1 lanes 0-15 = rows K=0,1 ; lanes 16-31 = rows K=2,3.
        const int   g  = anchorBase + l15;
        const float ax = anchors[g * 3 + 0];
        const float ay = anchors[g * 3 + 1];
        v2f bm;
        bm.x = (half == 0) ? (ax * ax + ay * ay) : (-2.0f * ay);
        bm.y = (half == 0) ? (-2.0f * ax)        : 1.0f;

        const float* annB = annotations + (size_t)b * kM * 4;
        float bestD = 3.0e38f;
        int   bestI = 0;

        // Preload all four annotation records branchlessly (clamped index).
        float4 rec[4];
        #pragma unroll
        for (int t = 0; t < 4; ++t) {
            const int m  = t * 16 + l15;
            const int mc = (m < kM) ? m : (kM - 1);
            rec[t] = *reinterpret_cast<const float4*>(annB + mc * 4);
        }

        #pragma unroll
        for (int t = 0; t < 4; ++t) {
            const int  m     = t * 16 + l15;
            const bool valid = (m < kM);
            const float mx = rec[t].x;
            const float my = rec[t].y;
            const float n2 = mx * mx + my * my;
            // ---- A fragment (16x4, M x K). Row m = annotation t*16 + l15.
            // Padded rows: (1,0 | 0,1e30) -> d2 = |a|^2 + 1e30, never wins.
            v2f am;
            am.x = (half == 0) ? 1.0f : (valid ? my : 0.0f);
            am.y = (half == 0) ? (valid ? mx : 0.0f)
                               : (valid ? n2 : 1.0e30f);

            v8f cacc = {};
            v8f d2 = __builtin_amdgcn_wmma_f32_16x16x4_f32(
                false, am, false, bm, (short)0, cacc, false, false);

            // D layout: column n = lane&15, row m = k + 8*(lane>>4)
            #pragma unroll
            for (int k = 0; k < 8; ++k) {
                const int   idx = t * 16 + half * 8 + k;
                const float v   = d2[k];
                if (v < bestD) { bestD = v; bestI = idx; }  // ascending idx in-lane
            }
        }

        // Merge lane halves (both correspond to anchor anchorBase + l15).
        const float oD = __shfl_xor(bestD, 16, 32);
        const int   oI = __shfl_xor(bestI, 16, 32);
        if (oD < bestD || (oD == bestD && oI < bestI)) { bestD = oD; bestI = oI; }

        if (half == 0) {
            // ---- per-anchor tail (lanes 0-15 only), anchor g == base + lane
            const float aal = anchors[g * 3 + 2];
            const float4 bestAnn =
                *reinterpret_cast<const float4*>(annB + bestI * 4);
            const float annX = bestAnn.x;
            const float annY = bestAnn.y;
            const float annA = bestAnn.z;
            const float annC = bestAnn.w;
            const float dal  = fabsf(aal - annA);

            const bool pos = (bestD <= kMaxPos2) && (dal <= kMaxAng);
            const bool neg = (bestD >= kNegPos2) || (dal >= kNegAng);

            const int iy = (int)rintf(ay);
            const int ix = (int)rintf(ax);
            const float damp = states[(size_t)b * kH * kW + iy * kW + ix] ? 1.0f : kDamp;

            const int code = pos ? (int)annC : (neg ? -2 : -1);
            codes[(size_t)b * kA + g] = code;
            damps[(size_t)b * kA + g] = damp;

            if (pos) {
                const size_t rb = ((size_t)b * kA + g) * 3;
                const float r0 = regressions[rb + 0];
                const float r1 = regressions[rb + 1];
                const float r2 = regressions[rb + 2];
                const float dx = fabsf((annX - ax) - r0);
                const float dy = fabsf((annY - ay) - r1);
                const float slx = (dx <= 1.0f / 9.0f) ? 4.5f * dx * dx : dx - 0.5f / 9.0f;
                const float sly = (dy <= 1.0f / 9.0f) ? 4.5f * dy * dy : dy - 0.5f / 9.0f;
                xyC  = (slx + sly) * damp;
                angC = fmaxf((fabsf((annA - aal) - r2) - 10.0f) / 5.0f, 0.0f) * damp;
                nposC = 1.0f;
            }
        }
    }

    // ---- block reduction of the three partials
    __shared__ float sh[3 * 256];
    sh[threadIdx.x]       = xyC;
    sh[256 + threadIdx.x] = angC;
    sh[512 + threadIdx.x] = nposC;
    __syncthreads();
    for (int s = 128; s > 0; s >>= 1) {
        if (threadIdx.x < s) {
            sh[threadIdx.x]       += sh[threadIdx.x + s];
            sh[256 + threadIdx.x] += sh[256 + threadIdx.x + s];
            sh[512 + threadIdx.x] += sh[512 + threadIdx.x + s];
        }
        __syncthreads();
    }
    if (threadIdx.x == 0) {
        xyPart  [(size_t)b * nb1 + blockIdx.x] = sh[0];
        angPart [(size_t)b * nb1 + blockIdx.x] = sh[256];
        nposPart[(size_t)b * nb1 + blockIdx.x] = sh[512];
    }
}

// ---------------------------------------------------------------------------
// Kernel 2: focal-loss streaming pass over classifications (256 MB).
// One thread per float4 chunk: thread -> (anchor = tid/20, chunk = tid%20),
// consecutive threads load consecutive float4s => fully coalesced.
// ---------------------------------------------------------------------------
__global__ __launch_bounds__(256)
void focal_kernel(const float* __restrict__ cls,     // (B,A,C)
                  const int*   __restrict__ codes,   // (B,A)
                  const float* __restrict__ damps,   // (B,A)
                  float* __restrict__ clsPart,       // (B,nb2)
                  int nb2)
{
    const int b   = blockIdx.y;
    const int tid = blockIdx.x * 256 + threadIdx.x;
    float acc = 0.0f;

    if (tid < kA * 20) {
        const int a     = tid / 20;
        const int chunk = tid - a * 20;
        const int code  = codes[(size_t)b * kA + a];
        if (code != -1) {   // ignore-anchors contribute zero
            const float damp = damps[(size_t)b * kA + a];
            const float4 v = *reinterpret_cast<const float4*>(
                cls + ((size_t)b * kA + a) * kC + chunk * 4);
            const float vv[4] = {v.x, v.y, v.z, v.w};
            #pragma unroll
            for (int j = 0; j < 4; ++j) {
                const int   cidx = chunk * 4 + j;
                const float c = fminf(fmaxf(vv[j], 1.0e-4f), 1.0f - 1.0e-4f);
                float l;
                if (cidx == code) {                 // target == 1 (pos class)
                    const float omc = 1.0f - c;
                    l = 0.95f * omc * omc * (-__logf(c));
                } else {                            // target == 0
                    l = 0.05f * c * c * (-__logf(1.0f - c));
                }
                acc += l;
            }
            acc *= damp;
        }
    }

    __shared__ float sh[256];
    sh[threadIdx.x] = acc;
    __syncthreads();
    for (int s = 128; s > 0; s >>= 1) {
        if (threadIdx.x < s) sh[threadIdx.x] += sh[threadIdx.x + s];
        __syncthreads();
    }
    if (threadIdx.x == 0) clsPart[(size_t)b * nb2 + blockIdx.x] = sh[0];
}

// ---------------------------------------------------------------------------
// Kernel 3: finalize. One block; one wave32 per batch sample.
// ---------------------------------------------------------------------------
__global__ __launch_bounds__(256)
void finalize_kernel(const float* __restrict__ clsPart, int nb2,
                     const float* __restrict__ xyPart,
                     const float* __restrict__ angPart,
                     const float* __restrict__ nposPart, int nb1,
                     float* __restrict__ out)
{
    __shared__ float sC[kB], sX[kB], sG[kB];
    const int b    = threadIdx.x >> 5;
    const int lane = threadIdx.x & 31;

    float cs = 0.0f, xs = 0.0f, gs = 0.0f, ns = 0.0f;
    for (int i = lane; i < nb2; i += 32) cs += clsPart[(size_t)b * nb2 + i];
    for (int i = lane; i < nb1; i += 32) {
        xs += xyPart  [(size_t)b * nb1 + i];
        gs += angPart [(size_t)b * nb1 + i];
        ns += nposPart[(size_t)b * nb1 + i];
    }
    for (int o = 16; o > 0; o >>= 1) {
        cs += __shfl_down(cs, o, 32);
        xs += __shfl_down(xs, o, 32);
        gs += __shfl_down(gs, o, 32);
        ns += __shfl_down(ns, o, 32);
    }
    if (lane == 0) {
        const float denom = fmaxf(ns, 1.0f);
        const bool  has   = ns > 0.0f;
        sC[b] = cs / denom;
        sX[b] = has ? xs / (2.0f * denom) : 0.0f;
        sG[b] = has ? gs / denom : 0.0f;
    }
    __syncthreads();
    if (threadIdx.x == 0) {
        float c = 0.0f, x = 0.0f, g = 0.0f;
        for (int i = 0; i < kB; ++i) { c += sC[i]; x += sX[i]; g += sG[i]; }
        out[0] = c / (float)kB;
        out[1] = x / (float)kB;
        out[2] = g / (float)kB;
    }
}

// ---------------------------------------------------------------------------
extern "C" void kernel_launch(void* const* d_in, const int* in_sizes, int n_in,
                              void* d_out, int out_size, void* d_ws, size_t ws_size,
                              hipStream_t stream) {
    const float*         cls  = (const float*)d_in[0];          // (B,A,C)
    const float*         reg  = (const float*)d_in[1];          // (B,A,3)
    const float*         anc  = (const float*)d_in[2];          // (1,A,3)
    const float*         ann  = (const float*)d_in[3];          // (B,M,4)
    const unsigned char* st   = (const unsigned char*)d_in[4];  // (B,1,H,W) bool

    const int nb1 = (kA / 16 + 7) / 8;                 // blocks of 128 anchors
    const int nb2 = (kA * 20 + 255) / 256;             // blocks of 256 float4-chunks

    char* ws = (char*)d_ws;
    size_t off = 0;
    int*   codes    = (int*)  (ws + off); off += (size_t)kB * kA * sizeof(int);
    float* damps    = (float*)(ws + off); off += (size_t)kB * kA * sizeof(float);
    float* xyPart   = (float*)(ws + off); off += (size_t)kB * nb1 * sizeof(float);
    float* angPart  = (float*)(ws + off); off += (size_t)kB * nb1 * sizeof(float);
    float* nposPart = (float*)(ws + off); off += (size_t)kB * nb1 * sizeof(float);
    float* clsPart  = (float*)(ws + off); off += (size_t)kB * nb2 * sizeof(float);
    (void)ws_size; (void)in_sizes; (void)n_in; (void)out_size;

    dim3 g1(nb1, kB);
    assign_wmma_kernel<<<g1, 256, 0, stream>>>(anc, ann, reg, st, codes, damps,
                                               xyPart, angPart, nposPart, nb1);

    dim3 g2(nb2, kB);
    focal_kernel<<<g2, 256, 0, stream>>>(cls, codes, damps, clsPart, nb2);

    finalize_kernel<<<1, 256, 0, stream>>>(clsPart, nb2, xyPart, angPart,
                                           nposPart, nb1, (float*)d_out);
}